// GraphSAGE_34514357191329
// MI455X (gfx1250) — compile-verified
//
#include <hip/hip_runtime.h>

// ---------------------------------------------------------------------------
// GraphSAGE (mean) x2 for MI455X / gfx1250, wave32.
//   layer: Y = relu(X @ Wself + (scatter_mean X) @ Wneigh + b)
// Phases: float atomics for edge aggregation (bandwidth/atomic bound),
//         FP32 WMMA (v_wmma_f32_16x16x4_f32) for the fused dual GEMM.
// ---------------------------------------------------------------------------

typedef float v2f __attribute__((ext_vector_type(2)));
typedef float v8f __attribute__((ext_vector_type(8)));

#define NFEAT 128  // IN_F == HID_F == 128

// ---------------- zero fill (float4 granularity) ---------------------------
__global__ __launch_bounds__(256) void zero_f32(float* __restrict__ p, long long n) {
  long long i = ((long long)blockIdx.x * blockDim.x + threadIdx.x) * 4;
  if (i + 3 < n) {
    *(float4*)(p + i) = make_float4(0.f, 0.f, 0.f, 0.f);
  } else {
    for (long long j = i; j < n; ++j) p[j] = 0.f;
  }
}

// ---------------- in-degree via float atomics ------------------------------
__global__ __launch_bounds__(256) void degree_kernel(const long long* __restrict__ dst,
                                                     float* __restrict__ deg, int n_edges) {
  int e = blockIdx.x * blockDim.x + threadIdx.x;
  if (e < n_edges) {
    unsafeAtomicAdd(&deg[(long long)dst[e]], 1.0f);  // global_atomic_add_f32
  }
}

// ---------------- edge scatter-add: agg[dst] += feat[src] ------------------
// One thread per (edge, 4-float chunk): float4 gather + 4 f32 atomics.
__global__ __launch_bounds__(256) void scatter_add_kernel(const float* __restrict__ feat,
                                                          const long long* __restrict__ src,
                                                          const long long* __restrict__ dst,
                                                          float* __restrict__ agg,
                                                          long long total /* n_edges*32 */) {
  long long idx = (long long)blockIdx.x * blockDim.x + threadIdx.x;
  if (idx >= total) return;
  int e = (int)(idx >> 5);          // 32 chunks of 4 floats per 128-wide row
  int c = (int)(idx & 31) << 2;
  long long s = src[e];
  long long d = dst[e];
  const float4 v = *(const float4*)(feat + s * NFEAT + c);
  float* p = agg + d * NFEAT + c;
  unsafeAtomicAdd(p + 0, v.x);
  unsafeAtomicAdd(p + 1, v.y);
  unsafeAtomicAdd(p + 2, v.z);
  unsafeAtomicAdd(p + 3, v.w);
}

// ---------------- fused SAGE GEMM + bias + ReLU ----------------------------
// Y[M,HOUT] = relu( Xs @ Wself + (AGG/deg) @ Wneigh + b )
// 256 threads = 8 waves. Each wave owns one 16x16 output tile.
//   NT = HOUT/16 column tiles, MB = 8/NT row sub-tiles, RM = 16*MB rows / WG.
// A-frag (16x4 f32, per ISA): lanes 0-15 -> M=lane,K={0,1}; lanes 16-31 -> M,K={2,3}.
// B-frag (4x16 f32): lanes 0-15 -> N=lane,K={0,1}; lanes 16-31 -> N,K={2,3}.
// C/D (16x16 f32, 8 VGPRs): lane<16 -> M=j, lane>=16 -> M=j+8; N=lane&15.
template <int KIN, int HOUT>
__global__ __launch_bounds__(256) void sage_gemm_relu(const float* __restrict__ X,
                                                      const float* __restrict__ AGG,
                                                      const float* __restrict__ DEG,
                                                      const float* __restrict__ Wself,
                                                      const float* __restrict__ Wneigh,
                                                      const float* __restrict__ bias,
                                                      float* __restrict__ Y) {
  constexpr int NT  = HOUT / 16;
  constexpr int MB  = 8 / NT;
  constexpr int RM  = 16 * MB;
  constexpr int LDP = KIN + 4;  // pad to spread LDS banks

  __shared__ float Xs[RM * LDP];
  __shared__ float Hn[RM * LDP];

  const int tid = threadIdx.x;
  const long long row0 = (long long)blockIdx.x * RM;

  // Cooperative staged load: X rows and mean-normalized aggregate rows.
  for (int idx = tid * 4; idx < RM * KIN; idx += 256 * 4) {
    int r = idx / KIN;
    int c = idx % KIN;
    long long g = (row0 + r) * KIN + c;
    float4 xv = *(const float4*)(X + g);
    float4 av = *(const float4*)(AGG + g);
    float inv = 1.0f / fmaxf(DEG[row0 + r], 1.0f);
    *(float4*)(&Xs[r * LDP + c]) = xv;
    float4 hv = make_float4(av.x * inv, av.y * inv, av.z * inv, av.w * inv);
    *(float4*)(&Hn[r * LDP + c]) = hv;
  }
  __syncthreads();

  const int wave  = tid >> 5;
  const int lane  = tid & 31;
  const int ntile = wave % NT;
  const int msub  = wave / NT;
  const int col0  = ntile * 16;
  const int c15   = lane & 15;
  const int kb    = (lane >> 4) * 2;     // K sub-offset for this half-wave
  const int ar    = msub * 16 + c15;     // A-frag row in LDS tile

  v8f acc = {};
  // Self term: Xs @ Wself
  for (int k0 = 0; k0 < KIN; k0 += 4) {
    v2f a, b;
    a.x = Xs[ar * LDP + k0 + kb];
    a.y = Xs[ar * LDP + k0 + kb + 1];
    b.x = Wself[(k0 + kb) * HOUT + col0 + c15];
    b.y = Wself[(k0 + kb + 1) * HOUT + col0 + c15];
    acc = __builtin_amdgcn_wmma_f32_16x16x4_f32(false, a, false, b, (short)0, acc, false, false);
  }
  // Neighbor term: Hn @ Wneigh
  for (int k0 = 0; k0 < KIN; k0 += 4) {
    v2f a, b;
    a.x = Hn[ar * LDP + k0 + kb];
    a.y = Hn[ar * LDP + k0 + kb + 1];
    b.x = Wneigh[(k0 + kb) * HOUT + col0 + c15];
    b.y = Wneigh[(k0 + kb + 1) * HOUT + col0 + c15];
    acc = __builtin_amdgcn_wmma_f32_16x16x4_f32(false, a, false, b, (short)0, acc, false, false);
  }

  // Epilogue: bias + ReLU, write D per the 8-VGPR C/D layout.
  const float bv = bias[col0 + c15];
  const int mhalf = (lane >> 4) * 8;
  for (int j = 0; j < 8; ++j) {
    int rr = msub * 16 + mhalf + j;
    float v = fmaxf(acc[j] + bv, 0.0f);
    Y[(row0 + rr) * HOUT + col0 + c15] = v;
  }
}

// ---------------------------------------------------------------------------
extern "C" void kernel_launch(void* const* d_in, const int* in_sizes, int n_in,
                              void* d_out, int out_size, void* d_ws, size_t ws_size,
                              hipStream_t stream) {
  const float*     feat = (const float*)d_in[0];       // [N,128] f32
  const long long* src  = (const long long*)d_in[1];   // [E] i64
  const long long* dst  = (const long long*)d_in[2];   // [E] i64
  const float*     Ws1  = (const float*)d_in[3];       // [128,128]
  const float*     Wn1  = (const float*)d_in[4];       // [128,128]
  const float*     b1   = (const float*)d_in[5];       // [128]
  const float*     Ws2  = (const float*)d_in[6];       // [128,64]
  const float*     Wn2  = (const float*)d_in[7];       // [128,64]
  const float*     b2   = (const float*)d_in[8];       // [64]

  const int n_nodes = in_sizes[0] / NFEAT;             // 100000 (16|32 divide it)
  const int n_edges = in_sizes[1];                     // 1600000

  // Workspace layout: agg[N*128] | deg[N] | h[N*128]  (~103 MB)
  float* agg = (float*)d_ws;
  float* deg = agg + (long long)n_nodes * NFEAT;
  float* h   = deg + n_nodes;

  const long long zAggDeg = (long long)n_nodes * (NFEAT + 1);
  const long long zAgg    = (long long)n_nodes * NFEAT;
  const long long total   = (long long)n_edges * 32;   // (edge, 4-float chunk) work-items

  // Layer 1
  zero_f32<<<(unsigned)((zAggDeg / 4 + 255) / 256 + 1), 256, 0, stream>>>(agg, zAggDeg);
  degree_kernel<<<(n_edges + 255) / 256, 256, 0, stream>>>(dst, deg, n_edges);
  scatter_add_kernel<<<(unsigned)((total + 255) / 256), 256, 0, stream>>>(feat, src, dst, agg, total);
  sage_gemm_relu<128, 128><<<n_nodes / 16, 256, 0, stream>>>(feat, agg, deg, Ws1, Wn1, b1, h);

  // Layer 2 (reuse agg buffer)
  zero_f32<<<(unsigned)((zAgg / 4 + 255) / 256 + 1), 256, 0, stream>>>(agg, zAgg);
  scatter_add_kernel<<<(unsigned)((total + 255) / 256), 256, 0, stream>>>(h, src, dst, agg, total);
  sage_gemm_relu<128, 64><<<n_nodes / 32, 256, 0, stream>>>(h, agg, deg, Ws2, Wn2, b2, (float*)d_out);
}